// NMS_2370821948166
// MI455X (gfx1250) — compile-verified
//
#include <hip/hip_runtime.h>
#include <stdint.h>

// ---------------------------------------------------------------------------
// Greedy NMS for MI455X (gfx1250, wave32).
//
//   d2(i,j) = ||a_i - b_j + eps||^2
//           = (||a_i||^2 + 2*eps*S_i + 3*eps^2)   [row term p]
//           + (||b_j||^2 - 2*eps*S_j)             [col term q]
//           - 2 * <a_i, b_j>                      [Gram via V_WMMA_F32_16X16X4_F32]
//
// Phase 1: rank-sort by score (descending, stable) -> sorted SoA + p/q terms.
// Phase 2: suppression bitmask matrix (16384 x 512 u32) via f32 WMMA + ballot.
// Phase 3: sequential greedy scan, one block, kept-mask in registers,
//          row prefetch overlapped with the ping-pong barrier.
// ---------------------------------------------------------------------------

#define EPSV 1e-6f
#define RAD2 4.0f   // NMS_RADIUS^2

typedef float v2f __attribute__((ext_vector_type(2)));
typedef float v8f __attribute__((ext_vector_type(8)));

// ---------------- Phase 1: rank (stable argsort by -score) -----------------
__global__ __launch_bounds__(256) void nms_rank_kernel(
    const float* __restrict__ nodes, const float* __restrict__ score,
    float* __restrict__ sx, float* __restrict__ sy, float* __restrict__ sz,
    float* __restrict__ pterm, float* __restrict__ qterm,
    int* __restrict__ orig, int n)
{
    __shared__ float stile[256];
    const int i = blockIdx.x * 256 + threadIdx.x;
    const float si = score[i];
    int rank = 0;
    for (int base = 0; base < n; base += 256) {
        stile[threadIdx.x] = score[base + threadIdx.x];
        __syncthreads();
        #pragma unroll 8
        for (int t = 0; t < 256; ++t) {
            const float sj = stile[t];
            const int j = base + t;
            // jnp.argsort(-s) stable: j before i iff s_j > s_i, or tie && j < i
            rank += (sj > si || (sj == si && j < i)) ? 1 : 0;
        }
        __syncthreads();
    }
    const float x = nodes[3 * i + 0];
    const float y = nodes[3 * i + 1];
    const float z = nodes[3 * i + 2];
    sx[rank] = x; sy[rank] = y; sz[rank] = z;
    const float n2 = x * x + y * y + z * z;
    const float S  = x + y + z;
    pterm[rank] = n2 + 2.0f * EPSV * S + 3.0f * EPSV * EPSV;
    qterm[rank] = n2 - 2.0f * EPSV * S;
    orig[rank]  = i;
}

// ------------- Phase 2: suppression bitmask via WMMA f32 16x16x4 -----------
// Block = 256 threads = 8 waves. blockIdx.y = row tile (16 rows),
// blockIdx.x*8 + wave = 32-column word index. Each wave: 2 WMMAs -> 16x32
// Gram patch -> threshold -> ballot-pack into 16 full u32 bitmask words.
__global__ __launch_bounds__(256) void nms_adj_kernel(
    const float* __restrict__ sx, const float* __restrict__ sy,
    const float* __restrict__ sz, const float* __restrict__ pterm,
    const float* __restrict__ qterm, uint32_t* __restrict__ bitmask, int n)
{
    const int rowWords = n >> 5;
    const int rowBase  = blockIdx.y << 4;
    const int lane     = threadIdx.x & 31;
    const int wave     = threadIdx.x >> 5;
    const int wordIdx  = (blockIdx.x << 3) + wave;
    const int colBase  = wordIdx << 5;

    __shared__ float pl[16];
    if (threadIdx.x < 16) pl[threadIdx.x] = pterm[rowBase + threadIdx.x];
    __syncthreads();

    const int  rl = lane & 15;
    const bool hi = lane >= 16;

    // A fragment (16x4, K: x,y,z,0): lanes<16 -> (K0,K1)=(x,y); lanes>=16 -> (K2,K3)=(z,0)
    const int ai = rowBase + rl;
    v2f a;
    a[0] = hi ? sz[ai] : sx[ai];
    a[1] = hi ? 0.0f   : sy[ai];

    // B fragments (4x16), mirrored layout, two adjacent column tiles
    const int j0 = colBase + rl;
    const int j1 = j0 + 16;
    v2f b0, b1;
    b0[0] = hi ? sz[j0] : sx[j0];
    b0[1] = hi ? 0.0f   : sy[j0];
    b1[0] = hi ? sz[j1] : sx[j1];
    b1[1] = hi ? 0.0f   : sy[j1];

    v8f c0 = {}; v8f c1 = {};
    c0 = __builtin_amdgcn_wmma_f32_16x16x4_f32(false, a, false, b0, (short)0, c0, false, false);
    c1 = __builtin_amdgcn_wmma_f32_16x16x4_f32(false, a, false, b1, (short)0, c1, false, false);

    const float q0 = qterm[colBase + rl];
    const float q1 = qterm[colBase + 16 + rl];

    // C/D layout: VGPR k, lanes 0-15 -> row k; lanes 16-31 -> row k+8; col = lane&15.
    uint32_t myWord = 0u;
    #pragma unroll
    for (int k = 0; k < 8; ++k) {
        const float p   = hi ? pl[k + 8] : pl[k];
        const bool pr0  = (p + q0 - 2.0f * c0[k]) <= RAD2;
        const bool pr1  = (p + q1 - 2.0f * c1[k]) <= RAD2;
        const uint32_t m0 = (uint32_t)__ballot(pr0); // [15:0]=row k, [31:16]=row k+8 (cols colBase..+15)
        const uint32_t m1 = (uint32_t)__ballot(pr1); // same, cols colBase+16..+31
        const uint32_t wlo = (m0 & 0xFFFFu) | (m1 << 16);          // row k, full 32-col word
        const uint32_t whi = (m0 >> 16)     | (m1 & 0xFFFF0000u);  // row k+8
        if (lane == k)     myWord = wlo;
        if (lane == k + 8) myWord = whi;
    }
    if (lane < 16)
        bitmask[(size_t)(rowBase + lane) * rowWords + wordIdx] = myWord;
}

// ------------- Phase 3: sequential greedy scan (single block) --------------
// Thread t owns kept-word t (sorted indices 32t..32t+31) in a register.
// Per candidate i: conflict = row_word & kept_word; block-OR via ping-pong
// LDS flag (2 barriers, race-free); only thread i>>5 updates its own word.
// Next row prefetched before the barriers so load latency overlaps sync.
__global__ __launch_bounds__(512) void nms_scan_kernel(
    const uint32_t* __restrict__ bitmask, const int* __restrict__ orig,
    float* __restrict__ out, int n)
{
    const int rowWords = n >> 5;
    const int tid = threadIdx.x;
    __shared__ int flags[2];
    __shared__ int total;
    if (tid == 0) { flags[0] = 0; flags[1] = 0; total = 0; }
    __syncthreads();

    uint32_t kept = 0u;
    uint32_t row  = bitmask[tid];            // row 0, word tid
    for (int i = 0; i < n; ++i) {
        uint32_t nxt = 0u;
        if (i + 1 < n) nxt = bitmask[(size_t)(i + 1) * rowWords + tid];
        const int idx = i & 1;
        if (row & kept)  flags[idx] = 1;     // any-conflict (benign multi-write)
        if (tid == 0)    flags[idx ^ 1] = 0; // reset other flag for next iter
        __syncthreads();
        if (flags[idx] == 0 && tid == (i >> 5))
            kept |= (1u << (i & 31));        // keep candidate i
        __syncthreads();
        row = nxt;
    }

    // Unsort mask + count.
    #pragma unroll
    for (int b = 0; b < 32; ++b) {
        const int r = (tid << 5) + b;
        out[orig[r]] = ((kept >> b) & 1u) ? 1.0f : 0.0f;
    }
    atomicAdd(&total, __popc(kept));
    __syncthreads();
    if (tid == 0) out[n] = (float)total;
}

// ---------------------------------------------------------------------------
extern "C" void kernel_launch(void* const* d_in, const int* in_sizes, int n_in,
                              void* d_out, int out_size, void* d_ws, size_t ws_size,
                              hipStream_t stream)
{
    const float* nodes = (const float*)d_in[0];
    const float* score = (const float*)d_in[1];
    const int n = in_sizes[1];               // 16384
    const int rowWords = n >> 5;             // 512

    // Workspace layout: [bitmask n*rowWords u32 = 32MB][sx][sy][sz][p][q][orig]
    uint32_t* bitmask = (uint32_t*)d_ws;
    float* sx    = (float*)(bitmask + (size_t)n * rowWords);
    float* sy    = sx + n;
    float* sz    = sy + n;
    float* pterm = sz + n;
    float* qterm = pterm + n;
    int*   orig  = (int*)(qterm + n);
    float* out   = (float*)d_out;

    nms_rank_kernel<<<n / 256, 256, 0, stream>>>(nodes, score, sx, sy, sz,
                                                 pterm, qterm, orig, n);
    dim3 gridB(rowWords / 8, n / 16);
    nms_adj_kernel<<<gridB, 256, 0, stream>>>(sx, sy, sz, pterm, qterm,
                                              bitmask, n);
    nms_scan_kernel<<<1, rowWords, 0, stream>>>(bitmask, orig, out, n);
}